// SpMergeAttentionLayer_88235808129632
// MI455X (gfx1250) — compile-verified
//
#include <hip/hip_runtime.h>
#include <hip/hip_bf16.h>

typedef __attribute__((ext_vector_type(2))) float v2f;
typedef __attribute__((ext_vector_type(8))) float v8f;

#define DOUTC 64
#define ALPHA_SLOPE 0.2f

// ---------------------------------------------------------------------------
// Kernel 1: Wr[r] = att[r,0]*basis[0] + att[r,1]*basis[1]   (2 x 64 x 64)
// ---------------------------------------------------------------------------
__global__ void wr_kernel(const float* __restrict__ att,
                          const float* __restrict__ basis,
                          float* __restrict__ Wr) {
    int i = blockIdx.x * blockDim.x + threadIdx.x;   // 0..8191
    if (i < 8192) {
        int r  = i >> 12;        // 0 or 1
        int io = i & 4095;       // index into 64x64
        Wr[i] = att[r * 2 + 0] * basis[io] + att[r * 2 + 1] * basis[4096 + io];
    }
}

// ---------------------------------------------------------------------------
// Kernel 2: h_pos = X @ Wr0, h_neg = X @ Wr1 using V_WMMA_F32_16X16X4_F32.
// Block = 128 threads = 4 waves; each wave computes a 16-row x 64-col strip
// of BOTH h_pos and h_neg: 16 k-steps x 4 n-tiles x 2 mats = 128 WMMAs/wave.
//
// W is stored in LDS *pair-interleaved*:  sW[r][ (k/2)*64 + n ]*2 + (k&1)
// so a lane's B fragment (W[k][n], W[k+1][n]) is one aligned ds_load_b64.
// The X tile rows are padded to 68 floats; (k even) A fragments are aligned
// v2f loads too.
//
// A fragment layout (f32 16x4): lanes 0-15 M=lane, K=(0,1); lanes 16-31
// M=lane-16, K=(2,3). B fragment symmetric with N=lane&15.
// C/D: VGPR v holds row M=v (lanes 0-15) / M=v+8 (lanes 16-31), N=lane&15.
// ---------------------------------------------------------------------------
__global__ __launch_bounds__(128) void gemm_h_kernel(
    const float* __restrict__ X,      // n x 64
    const float* __restrict__ Wr,     // 2 x 64 x 64 (row-major [k][n])
    float* __restrict__ hpos,         // n x 64
    float* __restrict__ hneg,         // n x 64
    int n) {
    __shared__ __align__(16) float sW[8192];        // 32 KB, pair-interleaved
    __shared__ __align__(16) float sX[4 * 16 * 68]; // 4 waves x 16 rows x 68 (pad)

    const int tid  = threadIdx.x;
    const int wave = tid >> 5;
    const int lane = tid & 31;

    // fill W interleaved: dst[((k>>1)*64 + n)*2 + (k&1)] = W[k][n]
    for (int i = tid; i < 8192; i += 128) {
        int r = i >> 12;
        int k = (i >> 6) & 63;
        int nn = i & 63;
        sW[(r << 12) + (((k >> 1) * 64 + nn) << 1) + (k & 1)] = Wr[i];
    }

    const int base = (blockIdx.x * 4 + wave) * 16;   // first global row of strip
    float* sx = &sX[wave * 16 * 68];

    // cooperative load of the 16x64 X tile (float4, coalesced), row clamp for tail
    for (int t = lane; t < 256; t += 32) {
        int row = t >> 4;
        int c4  = t & 15;
        int grow = base + row;
        if (grow >= n) grow = n - 1;                 // duplicate last row (stores guarded)
        float4 v = *(const float4*)(X + (size_t)grow * DOUTC + c4 * 4);
        *(float4*)(&sx[row * 68 + c4 * 4]) = v;      // 272B row stride keeps 16B align
    }
    __syncthreads();

    v8f accP[4] = {};
    v8f accN[4] = {};

    const int half = lane >> 4;          // 0 or 1
    const int mn   = lane & 15;

    for (int kk = 0; kk < 16; ++kk) {
        const int k  = kk * 4 + half * 2;   // even -> 8B aligned
        const int kp = kk * 2 + half;       // interleaved pair index 0..31
        const v2f a = *(const v2f*)&sx[mn * 68 + k];
#pragma unroll
        for (int t = 0; t < 4; ++t) {
            const int ncol = t * 16 + mn;
            const v2f bp = *(const v2f*)&sW[((kp * 64 + ncol) << 1)];
            const v2f bn = *(const v2f*)&sW[4096 + ((kp * 64 + ncol) << 1)];
            accP[t] = __builtin_amdgcn_wmma_f32_16x16x4_f32(
                false, a, false, bp, (short)0, accP[t], false, false);
            accN[t] = __builtin_amdgcn_wmma_f32_16x16x4_f32(
                false, a, false, bn, (short)0, accN[t], false, false);
        }
    }

    const int rofs = half * 8;
#pragma unroll
    for (int t = 0; t < 4; ++t) {
#pragma unroll
        for (int v = 0; v < 8; ++v) {
            const int grow = base + v + rofs;
            if (grow < n) {
                const size_t o = (size_t)grow * DOUTC + t * 16 + mn;
                hpos[o] = accP[t][v];
                hneg[o] = accN[t][v];
            }
        }
    }
}

// ---------------------------------------------------------------------------
// Kernel 3: per-node scalars  a = h . m_left, b = h . m_right  (pos & neg)
// One wave per node row; lane handles columns (lane, lane+32); shfl reduce.
// ---------------------------------------------------------------------------
__global__ void scalars_kernel(const float* __restrict__ hpos,
                               const float* __restrict__ hneg,
                               const float* __restrict__ mf,   // 128 floats
                               float* __restrict__ apos, float* __restrict__ bpos,
                               float* __restrict__ aneg, float* __restrict__ bneg,
                               int n) {
    const int lane = threadIdx.x & 31;
    const int wid  = (blockIdx.x * blockDim.x + threadIdx.x) >> 5;
    const int nw   = (gridDim.x * blockDim.x) >> 5;

    const float ml0 = mf[lane],      ml1 = mf[lane + 32];
    const float mr0 = mf[64 + lane], mr1 = mf[96 + lane];

    for (int i = wid; i < n; i += nw) {
        const size_t o = (size_t)i * DOUTC;
        const float hp0 = hpos[o + lane], hp1 = hpos[o + lane + 32];
        const float hn0 = hneg[o + lane], hn1 = hneg[o + lane + 32];
        float pa = hp0 * ml0 + hp1 * ml1;
        float pb = hp0 * mr0 + hp1 * mr1;
        float na = hn0 * ml0 + hn1 * ml1;
        float nb = hn0 * mr0 + hn1 * mr1;
#pragma unroll
        for (int off = 16; off > 0; off >>= 1) {
            pa += __shfl_xor(pa, off, 32);
            pb += __shfl_xor(pb, off, 32);
            na += __shfl_xor(na, off, 32);
            nb += __shfl_xor(nb, off, 32);
        }
        if (lane == 0) {
            apos[i] = pa; bpos[i] = pb;
            aneg[i] = na; bneg[i] = nb;
        }
    }
}

// ---------------------------------------------------------------------------
// Kernel 4a: zero degree array
// ---------------------------------------------------------------------------
__global__ void zero_kernel(float* __restrict__ p, int n) {
    int i = blockIdx.x * blockDim.x + threadIdx.x;
    if (i < n) p[i] = 0.0f;
}

// Kernel 4b: degree histogram over rows of both edge sets
__global__ void degree_kernel(const int* __restrict__ rows_pos,
                              const int* __restrict__ rows_neg,
                              int epos, int eneg, float* __restrict__ deg) {
    int i = blockIdx.x * blockDim.x + threadIdx.x;
    if (i < epos) {
        atomicAdd(&deg[rows_pos[i]], 1.0f);
    } else if (i < epos + eneg) {
        atomicAdd(&deg[rows_neg[i - epos]], 1.0f);
    }
}

// Kernel 4c: inv[i] = 1/sqrt(deg[i])
__global__ void invsqrt_kernel(const float* __restrict__ deg,
                               float* __restrict__ inv, int n) {
    int i = blockIdx.x * blockDim.x + threadIdx.x;
    if (i < n) inv[i] = rsqrtf(deg[i]);
}

// ---------------------------------------------------------------------------
// Kernel 5: initialize output with broadcast bias
// ---------------------------------------------------------------------------
__global__ void init_out_kernel(const float* __restrict__ bias,
                                float* __restrict__ out, long long total) {
    long long i = (long long)blockIdx.x * blockDim.x + threadIdx.x;
    if (i < total) out[i] = bias[i & 63];
}

// ---------------------------------------------------------------------------
// Kernel 6: edge aggregation. One wave per edge:
//   coef = sign * sigmoid(leaky_relu(a[r]+b[c])) * inv[r] * inv[c]
//   out[r,:] += coef * h[c,:]    (float atomics, 2 per lane)
// ---------------------------------------------------------------------------
__global__ void agg_kernel(const int* __restrict__ rows,
                           const int* __restrict__ cols, int E,
                           const float* __restrict__ a,
                           const float* __restrict__ b,
                           const float* __restrict__ inv,
                           const float* __restrict__ h,
                           float sign, float* __restrict__ out) {
    const int lane = threadIdx.x & 31;
    const int wid  = (blockIdx.x * blockDim.x + threadIdx.x) >> 5;
    const int nw   = (gridDim.x * blockDim.x) >> 5;

    for (int e = wid; e < E; e += nw) {
        const int r = rows[e];
        const int c = cols[e];
        float s  = a[r] + b[c];
        float lr = (s > 0.0f) ? s : ALPHA_SLOPE * s;
        float coef = 1.0f / (1.0f + expf(-lr));
        coef *= inv[r] * inv[c] * sign;

        const size_t oc  = (size_t)c * DOUTC;
        const size_t orr = (size_t)r * DOUTC;
        const float v0 = h[oc + lane];
        const float v1 = h[oc + lane + 32];
        atomicAdd(&out[orr + lane],      coef * v0);
        atomicAdd(&out[orr + lane + 32], coef * v1);
    }
}

// ---------------------------------------------------------------------------
extern "C" void kernel_launch(void* const* d_in, const int* in_sizes, int n_in,
                              void* d_out, int out_size, void* d_ws, size_t ws_size,
                              hipStream_t stream) {
    const float* node_reps = (const float*)d_in[0];  // n x 64
    const float* basis     = (const float*)d_in[1];  // 2 x 64 x 64
    const float* att       = (const float*)d_in[2];  // 2 x 2
    const float* mf        = (const float*)d_in[3];  // 1 x 128
    const float* bias      = (const float*)d_in[4];  // 1 x 64
    const int*   adj_pos   = (const int*)d_in[5];    // 2 x EPOS
    const int*   adj_neg   = (const int*)d_in[6];    // 2 x ENEG

    const int n    = in_sizes[0] / DOUTC;
    const int epos = in_sizes[5] / 2;
    const int eneg = in_sizes[6] / 2;

    float* out = (float*)d_out;
    float* ws  = (float*)d_ws;

    // workspace layout (floats)
    float* Wr   = ws;                               // 8192
    float* hpos = Wr + 8192;                        // n*64
    float* hneg = hpos + (size_t)n * DOUTC;         // n*64
    float* apos = hneg + (size_t)n * DOUTC;         // n
    float* bpos = apos + n;                         // n
    float* aneg = bpos + n;                         // n
    float* bneg = aneg + n;                         // n
    float* deg  = bneg + n;                         // n
    float* inv  = deg + n;                          // n

    // 1) Wr
    wr_kernel<<<32, 256, 0, stream>>>(att, basis, Wr);

    // 2) h_pos / h_neg via fp32 WMMA (4 waves/block, 64 rows/block)
    gemm_h_kernel<<<(n + 63) / 64, 128, 0, stream>>>(node_reps, Wr, hpos, hneg, n);

    // 3) per-node attention scalars
    scalars_kernel<<<1024, 256, 0, stream>>>(hpos, hneg, mf,
                                             apos, bpos, aneg, bneg, n);

    // 4) degrees -> 1/sqrt(deg)
    zero_kernel<<<(n + 255) / 256, 256, 0, stream>>>(deg, n);
    degree_kernel<<<(epos + eneg + 255) / 256, 256, 0, stream>>>(
        adj_pos, adj_neg, epos, eneg, deg);
    invsqrt_kernel<<<(n + 255) / 256, 256, 0, stream>>>(deg, inv, n);

    // 5) out = bias (broadcast)
    const long long total = (long long)n * DOUTC;
    init_out_kernel<<<(int)((total + 255) / 256), 256, 0, stream>>>(bias, out, total);

    // 6) aggregate: + positive edges, - negative edges
    agg_kernel<<<8192, 256, 0, stream>>>(adj_pos, adj_pos + epos, epos,
                                         apos, bpos, inv, hpos, 1.0f, out);
    agg_kernel<<<8192, 256, 0, stream>>>(adj_neg, adj_neg + eneg, eneg,
                                         aneg, bneg, inv, hneg, -1.0f, out);
}